// _SoftDTW_5832565588019
// MI455X (gfx1250) — compile-verified
//
#include <hip/hip_runtime.h>

#define SDTW_N 512
#define SDTW_M 512
#define BLK    128          // threads per block (4 wave32)
#define RPT    4            // contiguous rows per thread -> 512 rows
#define BIGF   1000000.0f
#define EPS_F  1e-9f
// softmin in base-2: b_i = r_i * (-1/(GAMMA*ln2));  softmin = -GAMMA*ln2*(log2(sum+eps)+bmax)
#define NEG_B_SCALE (-14.426950408889634f)   // -10 / ln(2)
#define GLN2        (0.06931471805599453f)   // 0.1 * ln(2)

static __device__ __forceinline__ void wait_asynccnt0() {
#if __has_builtin(__builtin_amdgcn_s_wait_asynccnt)
    __builtin_amdgcn_s_wait_asynccnt(0);
#else
    asm volatile("s_wait_asynccnt 0x0" ::: "memory");
#endif
}

static __device__ __forceinline__ float fast_exp2(float v) {
#if __has_builtin(__builtin_amdgcn_exp2f)
    return __builtin_amdgcn_exp2f(v);      // v_exp_f32 (base-2)
#else
    return exp2f(v);
#endif
}

static __device__ __forceinline__ float fast_log2(float v) {
#if __has_builtin(__builtin_amdgcn_logf)
    return __builtin_amdgcn_logf(v);       // v_log_f32 (base-2)
#else
    return __log2f(v);
#endif
}

__global__ __launch_bounds__(BLK) void sdtw_wavefront_kernel(
    const float* __restrict__ x, const float* __restrict__ y,
    float* __restrict__ partial) {
    // y row lives at ysh[512..1023]; ysh[0..511] is a guard region so the
    // per-cell LDS address 512 + (k - i - 1) never needs clamping.
    __shared__ __align__(16) float ysh[SDTW_M + 512];
    // double-buffered cross-thread exchange: R_k of each thread's last row
    __shared__ float xch[2][BLK];

    const int b = blockIdx.x;
    const int t = threadIdx.x;

    // ---- stage y[b,:] into LDS via CDNA5 async global->LDS DMA (ASYNCcnt) ----
    {
        const float* gsrc = y + (size_t)b * SDTW_M + t * 4;
        // generic pointers to LDS carry the LDS byte offset in their low 32 bits
        unsigned ldsoff = (unsigned)(unsigned long long)(void*)&ysh[512 + t * 4];
        unsigned long long gaddr = (unsigned long long)gsrc;
        asm volatile("global_load_async_to_lds_b128 %0, %1, off"
                     :: "v"(ldsoff), "v"(gaddr) : "memory");
        wait_asynccnt0();
    }

    // ---- per-thread x values (rows i = 4t+1 .. 4t+4 use x[4t .. 4t+3]) ----
    const float4 xv = *(const float4*)(x + (size_t)b * SDTW_N + t * 4);
    float xq[RPT] = {xv.x, xv.y, xv.z, xv.w};

    // ---- DP state in registers: R_{k-2}[i], R_{k-1}[i] for each owned row ----
    float Rm2[RPT], Rm1[RPT];
#pragma unroll
    for (int q = 0; q < RPT; ++q) { Rm2[q] = BIGF; Rm1[q] = BIGF; }
    // carry = R_{k-2}[left row]; at k=2 this is R_0[4t]: 0 for t==0 (row 0), else BIG
    float carry = (t == 0) ? 0.0f : BIGF;
    xch[1][t] = BIGF;          // R_1[last owned row] = BIG for the k=2 read
    __syncthreads();           // publishes ysh (post-asynccnt wait) and xch[1]

    const int base = 4 * t + 1;      // i of q==0
    const int yb   = 511 - base;     // ysh index = k + yb - q  (== 512 + (j-1))

#pragma unroll 2
    for (int k = 2; k <= SDTW_N + SDTW_M; ++k) {
        // left-neighbor values: one LDS word per thread per step
        const float l1 = (t == 0) ? BIGF : xch[(k - 1) & 1][t - 1]; // R_{k-1}[i-1], q==0
        const float l0 = carry;                                      // R_{k-2}[i-1], q==0
        carry = l1;

        float rk[RPT];
#pragma unroll
        for (int q = 0; q < RPT; ++q) {
            const float r0 = (q == 0) ? l0 : Rm2[q - 1];  // R_{k-2}[i-1]
            const float r1 = (q == 0) ? l1 : Rm1[q - 1];  // R_{k-1}[i-1]
            const float r2 = Rm1[q];                      // R_{k-1}[i]
            const float b0 = r0 * NEG_B_SCALE;
            const float b1 = r1 * NEG_B_SCALE;
            const float b2 = r2 * NEG_B_SCALE;
            const float bmax = fmaxf(fmaxf(b0, b1), b2);
            const float rsum = fast_exp2(b0 - bmax) + fast_exp2(b1 - bmax)
                             + fast_exp2(b2 - bmax);
            const float smin = -GLN2 * (fast_log2(rsum + EPS_F) + bmax);
            const float d = fabsf(xq[q] - ysh[k + yb - q]);   // guard region: no clamp
            const bool valid = ((unsigned)(k - (base + q) - 1) < (unsigned)SDTW_M);
            rk[q] = valid ? (d + smin) : BIGF;
        }
#pragma unroll
        for (int q = 0; q < RPT; ++q) { Rm2[q] = Rm1[q]; Rm1[q] = rk[q]; }
        xch[k & 1][t] = rk[RPT - 1];   // publish last owned row for thread t+1
        __syncthreads();               // one barrier per anti-diagonal
    }

    // R_{N+M}[N] lives in thread 127, row q=3
    if (t == BLK - 1)
        partial[b] = Rm1[RPT - 1] * (1.0f / (float)(SDTW_N + SDTW_M));
}

__global__ __launch_bounds__(256) void sdtw_reduce_kernel(
    const float* __restrict__ partial, float* __restrict__ out, int B) {
    __shared__ float s[256];
    const int t = threadIdx.x;
    float acc = 0.0f;
    for (int i = t; i < B; i += 256) acc += partial[i];
    s[t] = acc;
    __syncthreads();
    for (int o = 128; o > 0; o >>= 1) {
        if (t < o) s[t] += s[t + o];
        __syncthreads();
    }
    if (t == 0) out[0] = s[0] / (float)B;
}

extern "C" void kernel_launch(void* const* d_in, const int* in_sizes, int n_in,
                              void* d_out, int out_size, void* d_ws, size_t ws_size,
                              hipStream_t stream) {
    const float* x = (const float*)d_in[0];
    const float* y = (const float*)d_in[1];
    float* out = (float*)d_out;
    float* partial = (float*)d_ws;              // B floats of scratch
    const int B = in_sizes[0] / SDTW_N;         // 256 for the reference shapes

    sdtw_wavefront_kernel<<<B, BLK, 0, stream>>>(x, y, partial);
    sdtw_reduce_kernel<<<1, 256, 0, stream>>>(partial, out, B);
}